// GraphNeuralNetwork_81698867905110
// MI455X (gfx1250) — compile-verified
//
#include <hip/hip_runtime.h>

typedef __attribute__((ext_vector_type(2))) float v2f;
typedef __attribute__((ext_vector_type(8))) float v8f;

#define NNODES 50000
#define FEAT 64
#define GEMM_TILES (NNODES / 16) // 3125, exact

// ---------------- utility kernels ----------------

__global__ void zero_kernel(float* __restrict__ p, int n) {
  int i = blockIdx.x * blockDim.x + threadIdx.x;
  int stride = gridDim.x * blockDim.x;
  for (; i < n; i += stride) p[i] = 0.0f;
}

__global__ void degree_kernel(const int* __restrict__ col,
                              float* __restrict__ deg, int E) {
  int e = blockIdx.x * blockDim.x + threadIdx.x;
  if (e < E) unsafeAtomicAdd(&deg[col[e]], 1.0f);
}

__global__ void norm_kernel(const float* __restrict__ deg,
                            float* __restrict__ dis,
                            float* __restrict__ invc, int n) {
  int i = blockIdx.x * blockDim.x + threadIdx.x;
  if (i < n) {
    float d = deg[i];
    float dm = fmaxf(d, 1.0f);
    dis[i]  = d > 0.0f ? rsqrtf(dm) : 0.0f;
    invc[i] = d > 0.0f ? (1.0f / dm) : 0.0f; // scatter-mean divisor
  }
}

__global__ void edgew_kernel(const int* __restrict__ row,
                             const int* __restrict__ col,
                             const float* __restrict__ w,
                             const float* __restrict__ dis,
                             float* __restrict__ ew, int E) {
  int e = blockIdx.x * blockDim.x + threadIdx.x;
  if (e < E) ew[e] = w[e] * dis[row[e]] * dis[col[e]];
}

// gather x[row], scale by ew, scatter-add to s[col]; 16 threads/edge, float4 each
__global__ void scatter_kernel(const float* __restrict__ xin,
                               const int* __restrict__ row,
                               const int* __restrict__ col,
                               const float* __restrict__ ew,
                               float* __restrict__ s, int E) {
  int tid = blockIdx.x * blockDim.x + threadIdx.x;
  int e = tid >> 4;
  if (e >= E) return;
  int c = (tid & 15) * 4;
  int r = row[e];
  int d = col[e];
  float w = ew[e];
  const float4 xv = *(const float4*)(xin + (size_t)r * FEAT + c);
  float* dst = s + (size_t)d * FEAT + c;
  unsafeAtomicAdd(dst + 0, w * xv.x);   // non-returning global_atomic_add_f32, L2-resident
  unsafeAtomicAdd(dst + 1, w * xv.y);
  unsafeAtomicAdd(dst + 2, w * xv.z);
  unsafeAtomicAdd(dst + 3, w * xv.w);
}

// ---------------- WMMA fp32 GEMM: out = relu(invc*s @ W + b) ----------------
// Block = 128 threads = 4 waves; each wave owns ONE 16-row M-tile and computes
// all 64 output columns with 4 accumulators (A fragment reused across 4 WMMAs).
// V_WMMA_F32_16X16X4_F32 layouts (ISA 7.12.2):
//   A 16x4 : lane L holds row M=L%16; VGPR0/1 = K {2*hi, 2*hi+1} (hi = L/16)
//   B 4x16 : lane L holds col N=L%16; VGPR0/1 = K {2*hi, 2*hi+1}
//   C/D    : VGPR r, lane L -> row r + 8*hi, col L%16
template <int RELU>
__global__ __launch_bounds__(128) void gemm_wmma_kernel(
    const float* __restrict__ s, const float* __restrict__ invc,
    const float* __restrict__ W, const float* __restrict__ bias,
    float* __restrict__ out) {
  const int wv   = threadIdx.x >> 5;
  const int tile = blockIdx.x * 4 + wv;
  if (tile >= GEMM_TILES) return;      // wave-uniform exit; EXEC all-ones below
  const int lane = threadIdx.x & 31;
  const int hi   = lane >> 4;
  const int lq   = lane & 15;

  const int mrow = tile * 16 + lq;
  const float scale = invc[mrow];
  const float* arow = s + (size_t)mrow * FEAT;

  v8f acc0 = {}, acc1 = {}, acc2 = {}, acc3 = {};
#pragma unroll
  for (int kb = 0; kb < FEAT; kb += 4) {
    const int ka = kb + 2 * hi;
    v2f a;
    a.x = arow[ka] * scale;            // fold scatter-mean divide into A
    a.y = arow[ka + 1] * scale;
    const float* w0 = W + ka * FEAT + lq;        // W row-major [K][N]
    const float* w1 = W + (ka + 1) * FEAT + lq;
    v2f b0 = {w0[0],  w1[0]};
    v2f b1 = {w0[16], w1[16]};
    v2f b2 = {w0[32], w1[32]};
    v2f b3 = {w0[48], w1[48]};
    acc0 = __builtin_amdgcn_wmma_f32_16x16x4_f32(false, a, false, b0, (short)0, acc0, false, false);
    acc1 = __builtin_amdgcn_wmma_f32_16x16x4_f32(false, a, false, b1, (short)0, acc1, false, false);
    acc2 = __builtin_amdgcn_wmma_f32_16x16x4_f32(false, a, false, b2, (short)0, acc2, false, false);
    acc3 = __builtin_amdgcn_wmma_f32_16x16x4_f32(false, a, false, b3, (short)0, acc3, false, false);
  }

  // epilogue: bias (+ReLU) and store all four 16x16 tiles
  v8f accs[4] = {acc0, acc1, acc2, acc3};
#pragma unroll
  for (int t = 0; t < 4; ++t) {
    const int col = t * 16 + lq;
    const float bv = bias[col];
#pragma unroll
    for (int r = 0; r < 8; ++r) {
      const int m = r + 8 * hi;
      float v = accs[t][r] + bv;
      if (RELU) v = fmaxf(v, 0.0f);
      out[(size_t)(tile * 16 + m) * FEAT + col] = v;
    }
  }
}

// ---------------- launch ----------------

extern "C" void kernel_launch(void* const* d_in, const int* in_sizes, int n_in,
                              void* d_out, int out_size, void* d_ws, size_t ws_size,
                              hipStream_t stream) {
  (void)n_in; (void)out_size; (void)ws_size;
  const float* x  = (const float*)d_in[0];
  const int*   ei = (const int*)d_in[1];   // [2, E] flat
  const float* wE = (const float*)d_in[2];
  const float* W1 = (const float*)d_in[3];
  const float* b1 = (const float*)d_in[4];
  const float* W2 = (const float*)d_in[5];
  const float* b2 = (const float*)d_in[6];
  const float* W3 = (const float*)d_in[7];
  const float* b3 = (const float*)d_in[8];

  const int E = in_sizes[2];               // edge_weight element count
  const int* row = ei;
  const int* col = ei + E;

  float* ws   = (float*)d_ws;
  float* deg  = ws;                        // 50048
  float* dis  = deg  + 50048;
  float* invc = dis  + 50048;
  float* ew   = invc + 50048;              // E
  float* s    = ew   + ((E + 63) & ~63);   // 50000*64 scatter accumulator
  float* h1   = s    + (size_t)NNODES * FEAT;
  float* h2   = h1   + (size_t)NNODES * FEAT;
  float* outp = (float*)d_out;

  const int NF = NNODES * FEAT;
  auto cdiv = [](int a, int b) { return (a + b - 1) / b; };
  const int sc_blocks = cdiv(E * 16, 256);
  const int gemm_blocks = cdiv(GEMM_TILES, 4);

  zero_kernel<<<cdiv(NNODES, 256), 256, 0, stream>>>(deg, NNODES);
  degree_kernel<<<cdiv(E, 256), 256, 0, stream>>>(col, deg, E);
  norm_kernel<<<cdiv(NNODES, 256), 256, 0, stream>>>(deg, dis, invc, NNODES);
  edgew_kernel<<<cdiv(E, 256), 256, 0, stream>>>(row, col, wE, dis, ew, E);

  // layer 1
  zero_kernel<<<cdiv(NF, 256), 256, 0, stream>>>(s, NF);
  scatter_kernel<<<sc_blocks, 256, 0, stream>>>(x, row, col, ew, s, E);
  gemm_wmma_kernel<1><<<gemm_blocks, 128, 0, stream>>>(s, invc, W1, b1, h1);
  // layer 2
  zero_kernel<<<cdiv(NF, 256), 256, 0, stream>>>(s, NF);
  scatter_kernel<<<sc_blocks, 256, 0, stream>>>(h1, row, col, ew, s, E);
  gemm_wmma_kernel<1><<<gemm_blocks, 128, 0, stream>>>(s, invc, W2, b2, h2);
  // layer 3
  zero_kernel<<<cdiv(NF, 256), 256, 0, stream>>>(s, NF);
  scatter_kernel<<<sc_blocks, 256, 0, stream>>>(h2, row, col, ew, s, E);
  gemm_wmma_kernel<0><<<gemm_blocks, 128, 0, stream>>>(s, invc, W3, b3, outp);
}